// HELoss_22986664968850
// MI455X (gfx1250) — compile-verified
//
#include <hip/hip_runtime.h>

#define TPB 256
static constexpr int   kC    = 32000;              // classes per row
static constexpr int   kNF4  = kC / 4;             // 8000 float4 per row
static constexpr int   kFull = kNF4 / TPB;         // 31 full rounds of 256 float4
static constexpr int   kTail = kNF4 - kFull * TPB; // 64 leftover float4
static constexpr float kS    = 30.0f;              // scale hyperparameter
static constexpr float kL2E  = 1.4426950408889634f; // log2(e)

__device__ __forceinline__ float fast_exp2(float x) {
#if __has_builtin(__builtin_amdgcn_exp2f)
  return __builtin_amdgcn_exp2f(x);    // v_exp_f32
#else
  return __expf(x * 0.6931471805599453f);
#endif
}

// Issue one async global->LDS 128b copy per template step, sharing one global
// base VGPR pair + one LDS base VGPR; INST_OFFSET applies to both addresses
// (CDNA5 ISA), so only the 24-bit immediate varies.
template <int R, int END>
struct AsyncIssue {
  static __device__ __forceinline__ void run(unsigned lds, const float* gp) {
    asm volatile("global_load_async_to_lds_b128 %0, %1, off offset:%c2 th:TH_LOAD_NT"
                 :: "v"(lds), "v"(gp), "n"(R * TPB * 16) : "memory");
    AsyncIssue<R + 1, END>::run(lds, gp);
  }
};
template <int END>
struct AsyncIssue<END, END> {
  static __device__ __forceinline__ void run(unsigned, const float*) {}
};

// Pass 1 slab: wait until this wave's slab R has landed (its async issues
// complete IN ORDER, and pass 1 reads only data this wave itself loaded),
// then fold the slab into the running max. Threshold kFull-1-R is exact for
// 31-issue waves and conservative for the 32-issue (tail) waves.
template <int R>
struct Pass1Slab {
  static __device__ __forceinline__ void run(const float4* rowf4, int tid,
                                             float& m0, float& m1,
                                             float& m2, float& m3) {
    asm volatile("s_wait_asynccnt %c0" :: "n"(kFull - 1 - R) : "memory");
    const float4 v = rowf4[tid + R * TPB];
    m0 = fmaxf(m0, v.x); m1 = fmaxf(m1, v.y);
    m2 = fmaxf(m2, v.z); m3 = fmaxf(m3, v.w);
    Pass1Slab<R + 1>::run(rowf4, tid, m0, m1, m2, m3);
  }
};
template <>
struct Pass1Slab<kFull> {
  static __device__ __forceinline__ void run(const float4*, int,
                                             float&, float&, float&, float&) {}
};

// One workgroup per row. Stream the 128KB row into LDS with gfx1250 async
// global->LDS copies (ASYNCcnt) while pass 1 consumes slabs as they arrive;
// then pass 2 does one v_exp_f32 per element from LDS.
__global__ __launch_bounds__(TPB) void heloss_row_kernel(
    const float* __restrict__ logits, const int* __restrict__ labels,
    const float* __restrict__ cm, float* __restrict__ rowloss, int nrows)
{
  __shared__ __align__(16) float row[kC];     // 128000 bytes
  __shared__ float redm[TPB / 32];
  __shared__ float reds[TPB / 32];

  const int rowi = blockIdx.x;
  if (rowi >= nrows) return;
  const int tid = threadIdx.x;
  const float* g = logits + (size_t)rowi * kC;

  // ---- Issue all async slab copies up front (deep ASYNCcnt pipeline) ----
  {
    const unsigned ldsbase = (unsigned)(unsigned long long)(&row[4 * tid]);
    const float*   gbase   = g + 4 * tid;
    AsyncIssue<0, kFull>::run(ldsbase, gbase);          // 31 full slabs
    if (tid < kTail) {                                  // 64-float4 tail slab
      asm volatile("global_load_async_to_lds_b128 %0, %1, off offset:%c2 th:TH_LOAD_NT"
                   :: "v"(ldsbase), "v"(gbase), "n"(kFull * TPB * 16) : "memory");
    }
  }
  const int   L   = labels[rowi];
  const float scm = kS * cm[0];

  const float4* rowf4 = reinterpret_cast<const float4*>(row);

  // ---- Pass 1 (pipelined with the copies): max of RAW logits.
  //      S>0 => max(S*x) = S*max(x); label column folded in afterwards as an
  //      upper bound: M = max(S*maxraw, S*x_y - scm).
  float m0 = -3.0e38f, m1 = -3.0e38f, m2 = -3.0e38f, m3 = -3.0e38f;
  Pass1Slab<0>::run(rowf4, tid, m0, m1, m2, m3);
  if (tid < kTail) {                      // asynccnt already waited to 0 at R=30
    const float4 v = rowf4[tid + kFull * TPB];
    m0 = fmaxf(m0, v.x); m1 = fmaxf(m1, v.y);
    m2 = fmaxf(m2, v.z); m3 = fmaxf(m3, v.w);
  }
  float m = fmaxf(fmaxf(m0, m1), fmaxf(m2, m3));
  #pragma unroll
  for (int off = 16; off > 0; off >>= 1)
    m = fmaxf(m, __shfl_xor(m, off, 32));
  if ((tid & 31) == 0) redm[tid >> 5] = m;
  __syncthreads();   // all waves reached here => all async writes visible too
  float mraw = redm[0];
  #pragma unroll
  for (int w = 1; w < TPB / 32; ++w) mraw = fmaxf(mraw, redm[w]);

  const float xy = row[L];                       // raw label logit (LDS bcast)
  const float M  = fmaxf(kS * mraw, kS * xy - scm);
  const float c0 = kS * kL2E;                    // exp(S*x - M) = exp2(c0*x + c1)
  const float c1 = -M * kL2E;

  // ---- Pass 2: sum exp over UNMODIFIED row: 1 fma + 1 v_exp_f32 + 1 add /elem
  float s0 = 0.0f, s1 = 0.0f, s2 = 0.0f, s3 = 0.0f;
  #pragma unroll
  for (int r = 0; r < kFull; ++r) {
    const float4 v = rowf4[tid + r * TPB];
    s0 += fast_exp2(fmaf(v.x, c0, c1));
    s1 += fast_exp2(fmaf(v.y, c0, c1));
    s2 += fast_exp2(fmaf(v.z, c0, c1));
    s3 += fast_exp2(fmaf(v.w, c0, c1));
  }
  if (tid < kTail) {
    const float4 v = rowf4[tid + kFull * TPB];
    s0 += fast_exp2(fmaf(v.x, c0, c1));
    s1 += fast_exp2(fmaf(v.y, c0, c1));
    s2 += fast_exp2(fmaf(v.z, c0, c1));
    s3 += fast_exp2(fmaf(v.w, c0, c1));
  }
  float s = (s0 + s1) + (s2 + s3);
  #pragma unroll
  for (int off = 16; off > 0; off >>= 1)
    s += __shfl_xor(s, off, 32);
  if ((tid & 31) == 0) reds[tid >> 5] = s;
  __syncthreads();

  if (tid == 0) {
    float st = 0.0f;
    #pragma unroll
    for (int w = 0; w < TPB / 32; ++w) st += reds[w];
    // Swap label column: remove exp(S*x_y - M), add exp(S*x_y - scm - M).
    const float num   = kS * xy - scm;           // S*(x_label - cm)
    const float s_mod = st - fast_exp2(fmaf(xy, c0, c1))
                           + fast_exp2((num - M) * kL2E);
    const float lse = M + __logf(s_mod);
    rowloss[rowi] = lse - num;                   // -(num - lse)
  }
}

// Deterministic final mean over the per-row losses.
__global__ __launch_bounds__(TPB) void heloss_reduce_kernel(
    const float* __restrict__ rowloss, float* __restrict__ out, int nrows)
{
  __shared__ float red[TPB / 32];
  float s = 0.0f;
  for (int i = threadIdx.x; i < nrows; i += TPB) s += rowloss[i];
  #pragma unroll
  for (int off = 16; off > 0; off >>= 1) s += __shfl_xor(s, off, 32);
  if ((threadIdx.x & 31) == 0) red[threadIdx.x >> 5] = s;
  __syncthreads();
  if (threadIdx.x == 0) {
    float t = 0.0f;
    #pragma unroll
    for (int w = 0; w < TPB / 32; ++w) t += red[w];
    out[0] = t / (float)nrows;
  }
}

extern "C" void kernel_launch(void* const* d_in, const int* in_sizes, int n_in,
                              void* d_out, int out_size, void* d_ws, size_t ws_size,
                              hipStream_t stream)
{
  const float* logits = (const float*)d_in[0];
  const int*   labels = (const int*)d_in[1];
  const float* cm     = (const float*)d_in[2];
  float*       out    = (float*)d_out;
  const int    nrows  = in_sizes[1];      // 8192
  float* rowloss = (float*)d_ws;          // nrows floats of scratch

  heloss_row_kernel<<<nrows, TPB, 0, stream>>>(logits, labels, cm, rowloss, nrows);
  heloss_reduce_kernel<<<1, TPB, 0, stream>>>(rowloss, out, nrows);
}